// ModalityTower_14431090115244
// MI455X (gfx1250) — compile-verified
//
#include <hip/hip_runtime.h>
#include <math.h>

#define L_    4
#define D_    512
#define FF_   1024
#define E_    6
#define H_    8
#define TOPK_ 2
#define B_    8
#define T_    1024
#define HD_   64
#define M_    (B_ * T_)      /* 8192 tokens */
#define F_    (T_ / 2 + 1)   /* 513 rfft bins */

typedef __attribute__((ext_vector_type(2))) float v2f;
typedef __attribute__((ext_vector_type(8))) float v8f;

#define FLAG_GELU  1
#define FLAG_ACC   2
#define FLAG_SCALE 4

#define BM  64
#define BN  128
#define BK  32
#define PAD 36   /* 16B-aligned rows; gcd(36,64)=4 -> conflict-free 16-lane columns */

__device__ __forceinline__ float gelu_exact(float x) {
    return 0.5f * x * (1.0f + erff(x * 0.70710678118654752f));
}

// ---------------------------------------------------------------------------
// fp32 WMMA GEMM:  C[M,N] = act(A[M,K] @ op(B) + bias) * scale (+ C)
//   transB==0 -> B is (N,K) row-major (C = A @ B^T);  transB==1 -> B[k*ldb+n]
//   batched over blockIdx.z via element strides (strideA may be 0 = shared A)
// Block tile 64x128, K-tile 32, 256 threads = 8 waves, each wave a 32x32
// quadrant (4 accumulators): 2 A-frags + 2 B-frags per 4 WMMAs. LDS tiles are
// double-buffered; next panel's b128 global loads are issued before the WMMA
// loop and stored after it, hiding VMEM latency behind matrix math.
// ---------------------------------------------------------------------------
__global__ __launch_bounds__(256) void gemm_f32_wmma(
    const float* __restrict__ A, const float* __restrict__ B,
    const float* __restrict__ bias, const float* __restrict__ scale,
    float* __restrict__ C,
    int M, int N, int K, int lda, int ldb, int ldc,
    long long strideA, long long strideB, long long strideC, long long strideS,
    int sstride, int transB, int flags)
{
    __shared__ float As[2][BM][PAD];
    __shared__ float Bs[2][BN][PAD];
    const int bz = blockIdx.z;
    A += (long long)bz * strideA;
    B += (long long)bz * strideB;
    C += (long long)bz * strideC;
    if (scale) scale += (long long)bz * strideS;

    const int m0 = blockIdx.y * BM;
    const int n0 = blockIdx.x * BN;
    const int tid  = threadIdx.x;
    const int lane = tid & 31;
    const int wave = tid >> 5;
    const int wrow = (wave & 1) * 32;   // wave's 32-row strip
    const int wcol = (wave >> 1) * 32;  // wave's 32-col strip

    const bool alignA = ((lda & 3) == 0);
    const bool alignB = ((ldb & 3) == 0);

    v8f acc00 = {}, acc01 = {}, acc10 = {}, acc11 = {};

    // staged registers: A 64x32/256thr = 2 float4, B 128x32/256thr = 4 float4
    float4 aR[2], bR[4];

    auto loadA = [&](int kt) {
        #pragma unroll
        for (int u = 0; u < 2; ++u) {
            int g  = tid + u * 256;       // 512 groups of 4
            int r  = g >> 3;
            int c4 = (g & 7) << 2;
            int gm = m0 + r, gk = kt + c4;
            float4 v = {0.f, 0.f, 0.f, 0.f};
            if (gm < M) {
                const float* p = A + (long long)gm * lda + gk;
                if (alignA && gk + 3 < K) v = *(const float4*)p;
                else {
                    if (gk     < K) v.x = p[0];
                    if (gk + 1 < K) v.y = p[1];
                    if (gk + 2 < K) v.z = p[2];
                    if (gk + 3 < K) v.w = p[3];
                }
            }
            aR[u] = v;
        }
    };
    auto storeA = [&](int buf) {
        #pragma unroll
        for (int u = 0; u < 2; ++u) {
            int g  = tid + u * 256;
            int r  = g >> 3;
            int c4 = (g & 7) << 2;
            *(float4*)&As[buf][r][c4] = aR[u];
        }
    };
    auto loadB = [&](int kt) {
        #pragma unroll
        for (int u = 0; u < 4; ++u) {
            int g = tid + u * 256;        // 1024 groups of 4
            float4 v = {0.f, 0.f, 0.f, 0.f};
            if (!transB) {
                int r  = g >> 3;          // n-row
                int c4 = (g & 7) << 2;    // k
                int gn = n0 + r, gk = kt + c4;
                if (gn < N) {
                    const float* p = B + (long long)gn * ldb + gk;
                    if (alignB && gk + 3 < K) v = *(const float4*)p;
                    else {
                        if (gk     < K) v.x = p[0];
                        if (gk + 1 < K) v.y = p[1];
                        if (gk + 2 < K) v.z = p[2];
                        if (gk + 3 < K) v.w = p[3];
                    }
                }
            } else {
                int k  = g >> 5;          // 32 n-groups per k
                int n4 = (g & 31) << 2;
                int gk = kt + k, gn = n0 + n4;
                if (gk < K) {
                    const float* p = B + (long long)gk * ldb + gn;
                    if (alignB && gn + 3 < N) v = *(const float4*)p;
                    else {
                        if (gn     < N) v.x = p[0];
                        if (gn + 1 < N) v.y = p[1];
                        if (gn + 2 < N) v.z = p[2];
                        if (gn + 3 < N) v.w = p[3];
                    }
                }
            }
            bR[u] = v;
        }
    };
    auto storeB = [&](int buf) {
        #pragma unroll
        for (int u = 0; u < 4; ++u) {
            int g = tid + u * 256;
            if (!transB) {
                int r  = g >> 3;
                int c4 = (g & 7) << 2;
                *(float4*)&Bs[buf][r][c4] = bR[u];
            } else {
                int k  = g >> 5;
                int n4 = (g & 31) << 2;
                Bs[buf][n4 + 0][k] = bR[u].x;
                Bs[buf][n4 + 1][k] = bR[u].y;
                Bs[buf][n4 + 2][k] = bR[u].z;
                Bs[buf][n4 + 3][k] = bR[u].w;
            }
        }
    };

    // ISA 7.12.2: lanes 0-15 hold K={0,1}, lanes 16-31 hold K={2,3}
    const int lr    = lane & 15;
    const int khalf = (lane >> 4) << 1;

    loadA(0); loadB(0);
    storeA(0); storeB(0);
    __syncthreads();

    const int nk = (K + BK - 1) / BK;
    for (int t = 0; t < nk; ++t) {
        const int cur = t & 1, nxt = cur ^ 1;
        const bool more = (t + 1 < nk);
        if (more) { loadA((t + 1) * BK); loadB((t + 1) * BK); }
        #pragma unroll
        for (int kk = 0; kk < BK; kk += 4) {
            v2f a0, a1, b0, b1;
            a0[0] = As[cur][wrow + lr][kk + khalf];
            a0[1] = As[cur][wrow + lr][kk + khalf + 1];
            a1[0] = As[cur][wrow + 16 + lr][kk + khalf];
            a1[1] = As[cur][wrow + 16 + lr][kk + khalf + 1];
            b0[0] = Bs[cur][wcol + lr][kk + khalf];
            b0[1] = Bs[cur][wcol + lr][kk + khalf + 1];
            b1[0] = Bs[cur][wcol + 16 + lr][kk + khalf];
            b1[1] = Bs[cur][wcol + 16 + lr][kk + khalf + 1];
            acc00 = __builtin_amdgcn_wmma_f32_16x16x4_f32(false, a0, false, b0,
                                                          (short)0, acc00, false, false);
            acc01 = __builtin_amdgcn_wmma_f32_16x16x4_f32(false, a0, false, b1,
                                                          (short)0, acc01, false, false);
            acc10 = __builtin_amdgcn_wmma_f32_16x16x4_f32(false, a1, false, b0,
                                                          (short)0, acc10, false, false);
            acc11 = __builtin_amdgcn_wmma_f32_16x16x4_f32(false, a1, false, b1,
                                                          (short)0, acc11, false, false);
        }
        if (more) { storeA(nxt); storeB(nxt); }
        __syncthreads();
    }

    // C/D layout: lane<16 -> rows 0..7, lane>=16 -> rows 8..15; N = lane&15
    #pragma unroll
    for (int mi = 0; mi < 2; ++mi) {
        const int gmBase = m0 + wrow + mi * 16 + ((lane >> 4) << 3);
        #pragma unroll
        for (int ni = 0; ni < 2; ++ni) {
            const int gn = n0 + wcol + ni * 16 + (lane & 15);
            if (gn >= N) continue;
            v8f acc = mi == 0 ? (ni == 0 ? acc00 : acc01)
                              : (ni == 0 ? acc10 : acc11);
            const float bval = bias ? bias[gn] : 0.f;
            #pragma unroll
            for (int i = 0; i < 8; ++i) {
                int gm = gmBase + i;
                if (gm >= M) continue;
                float v = acc[i] + bval;
                if (flags & FLAG_GELU)  v = gelu_exact(v);
                if (flags & FLAG_SCALE) v *= scale[(long long)gm * sstride];
                long long ci = (long long)gm * ldc + gn;
                if (flags & FLAG_ACC) C[ci] += v;
                else                  C[ci]  = v;
            }
        }
    }
}

// ---------------------------------------------------------------------------
// LayerNorm over D=512, one block per row
// ---------------------------------------------------------------------------
__global__ __launch_bounds__(256) void ln_kernel(const float* __restrict__ x,
        const float* __restrict__ w, const float* __restrict__ b,
        float* __restrict__ o)
{
    __shared__ float red[256];
    const int row = blockIdx.x;
    const int tid = threadIdx.x;
    const float* xr = x + (long long)row * D_;
    float v0 = xr[tid], v1 = xr[tid + 256];
    red[tid] = v0 + v1;
    __syncthreads();
    for (int off = 128; off > 0; off >>= 1) {
        if (tid < off) red[tid] += red[tid + off];
        __syncthreads();
    }
    float mean = red[0] / (float)D_;
    __syncthreads();
    float d0 = v0 - mean, d1 = v1 - mean;
    red[tid] = d0 * d0 + d1 * d1;
    __syncthreads();
    for (int off = 128; off > 0; off >>= 1) {
        if (tid < off) red[tid] += red[tid + off];
        __syncthreads();
    }
    float inv = rsqrtf(red[0] / (float)D_ + 1e-5f);
    float* orow = o + (long long)row * D_;
    orow[tid]       = d0 * inv * w[tid]       + b[tid];
    orow[tid + 256] = d1 * inv * w[tid + 256] + b[tid + 256];
}

// ---------------------------------------------------------------------------
// Attention: one wave per (b,h,q). Scores in LDS, float4 dot products,
// wave32 shfl reductions.
// ---------------------------------------------------------------------------
__global__ __launch_bounds__(256) void attn_kernel(const float* __restrict__ qkv,
        float* __restrict__ o, const float* __restrict__ log_alpha, int use_bias)
{
    __shared__ float sc[8][T_];
    __shared__ __align__(16) float qs[8][HD_];
    const int wave = threadIdx.x >> 5;
    const int lane = threadIdx.x & 31;
    const int wid  = blockIdx.x * 8 + wave;
    const int q = wid % T_;
    const int h = (wid / T_) % H_;
    const int b = wid / (T_ * H_);
    const float alpha = use_bias ? expf(log_alpha[0]) : 0.0f;
    const long long row3 = 3LL * D_;
    const long long base = ((long long)b * T_) * row3;

    for (int d = lane; d < HD_; d += 32)
        qs[wave][d] = qkv[base + (long long)q * row3 + h * HD_ + d];
    const float4* q4 = (const float4*)qs[wave];

    float mx = -1e30f;
    for (int k = lane; k < T_; k += 32) {
        const float4* kr4 = (const float4*)(qkv + base + (long long)k * row3
                                            + D_ + h * HD_);
        float s = 0.f;
        #pragma unroll
        for (int d4 = 0; d4 < HD_ / 4; ++d4) {
            float4 qv = q4[d4], kv = kr4[d4];
            s += qv.x * kv.x + qv.y * kv.y + qv.z * kv.z + qv.w * kv.w;
        }
        s *= 0.125f;                                  // 1/sqrt(64)
        if (use_bias) s -= alpha * fabsf((float)(q - k));
        sc[wave][k] = s;
        mx = fmaxf(mx, s);
    }
    for (int off = 16; off > 0; off >>= 1) mx = fmaxf(mx, __shfl_xor(mx, off, 32));
    float sum = 0.f;
    for (int k = lane; k < T_; k += 32) {
        float e = expf(sc[wave][k] - mx);
        sc[wave][k] = e;
        sum += e;
    }
    for (int off = 16; off > 0; off >>= 1) sum += __shfl_xor(sum, off, 32);
    const float inv = 1.0f / sum;
    float o0 = 0.f, o1 = 0.f;
    for (int k = 0; k < T_; ++k) {
        float e = sc[wave][k];
        const float* vr = qkv + base + (long long)k * row3 + 2 * D_ + h * HD_;
        o0 += e * vr[lane];
        o1 += e * vr[lane + 32];
    }
    const long long orow = ((long long)b * T_ + q) * D_ + h * HD_;
    o[orow + lane]      = o0 * inv;
    o[orow + lane + 32] = o1 * inv;
}

// ---------------------------------------------------------------------------
// Router: softmax over E=6, top-2 weights into dense gate, sigmoid gate,
// aux-loss partial sums (probs sum + top-2 counts) via atomics.
// ---------------------------------------------------------------------------
__global__ void router_kernel(const float* __restrict__ normed,
        const float* __restrict__ rw, const float* __restrict__ rb,
        const float* __restrict__ gw, const float* __restrict__ gb,
        float* __restrict__ gate, float* __restrict__ sgate,
        float* __restrict__ aux)
{
    int m = blockIdx.x * blockDim.x + threadIdx.x;
    if (m >= M_) return;
    const float* xr = normed + (long long)m * D_;
    float acc[E_] = {0.f, 0.f, 0.f, 0.f, 0.f, 0.f};
    float accg = 0.f;
    for (int d = 0; d < D_; ++d) {
        float xv = xr[d];
        #pragma unroll
        for (int e = 0; e < E_; ++e) acc[e] += xv * rw[e * D_ + d];
        accg += xv * gw[d];
    }
    float mx = -1e30f;
    #pragma unroll
    for (int e = 0; e < E_; ++e) { acc[e] += rb[e]; mx = fmaxf(mx, acc[e]); }
    float p[E_], se = 0.f;
    #pragma unroll
    for (int e = 0; e < E_; ++e) { p[e] = expf(acc[e] - mx); se += p[e]; }
    const float invs = 1.0f / se;
    #pragma unroll
    for (int e = 0; e < E_; ++e) p[e] *= invs;
    int i0 = 0;
    #pragma unroll
    for (int e = 1; e < E_; ++e) if (p[e] > p[i0]) i0 = e;
    int i1 = (i0 == 0) ? 1 : 0;
    #pragma unroll
    for (int e = 0; e < E_; ++e) if (e != i0 && p[e] > p[i1]) i1 = e;
    const float wsum = p[i0] + p[i1];
    #pragma unroll
    for (int e = 0; e < E_; ++e) gate[(long long)m * E_ + e] = 0.f;
    gate[(long long)m * E_ + i0] = p[i0] / wsum;
    gate[(long long)m * E_ + i1] = p[i1] / wsum;
    sgate[m] = 1.0f / (1.0f + expf(-(accg + gb[0])));
    #pragma unroll
    for (int e = 0; e < E_; ++e) atomicAdd(&aux[e], p[e]);
    atomicAdd(&aux[E_ + i0], 1.0f);
    atomicAdd(&aux[E_ + i1], 1.0f);
}

// depthwise conv (k=3, pad=1) + residual: cin = x + dwconv(x, k)
__global__ void dwconv_kernel(const float* __restrict__ xin,
        const float* __restrict__ kw, float* __restrict__ cout, int n)
{
    int i = blockIdx.x * blockDim.x + threadIdx.x;
    if (i >= n) return;
    int d = i % D_;
    int t = (i / D_) % T_;
    float c   = xin[i];
    float lft = (t > 0)      ? xin[i - D_] : 0.f;
    float rgt = (t < T_ - 1) ? xin[i + D_] : 0.f;
    cout[i] = c + lft * kw[d * 3 + 0] + c * kw[d * 3 + 1] + rgt * kw[d * 3 + 2];
}

// DFT bases. Forward: Re = cosF @ x, Im = sinF @ x (sinF holds -sin).
// Inverse (FFTW c2r: Im of DC/Nyquist ignored, interior bins weighted 2):
//   x[t] = sum_f icos[t,f]*Re'[f] + isin[t,f]*Im'[f]
__global__ void basis_kernel(float* __restrict__ cosF, float* __restrict__ sinF,
                             float* __restrict__ icos, float* __restrict__ isin)
{
    int i = blockIdx.x * blockDim.x + threadIdx.x;
    if (i >= F_ * T_) return;
    int f = i / T_;
    int t = i % T_;
    long long r = ((long long)f * t) % T_;
    double ang = 6.283185307179586476925286766559 * (double)r / (double)T_;
    float c = (float)cos(ang), s = (float)sin(ang);
    cosF[i] = c;
    sinF[i] = -s;
    int edge = (f == 0 || f == F_ - 1);
    float w = edge ? 1.0f : 2.0f;
    icos[t * F_ + f] = w * c / (float)T_;
    isin[t * F_ + f] = edge ? 0.f : (-w * s / (float)T_);
}

__global__ void copy_kernel(const float* __restrict__ a, float* __restrict__ o, int n)
{
    int i = blockIdx.x * blockDim.x + threadIdx.x;
    if (i < n) o[i] = a[i];
}
__global__ void add_kernel(float* __restrict__ o, const float* __restrict__ a, int n)
{
    int i = blockIdx.x * blockDim.x + threadIdx.x;
    if (i < n) o[i] += a[i];
}
__global__ void zero_kernel(float* __restrict__ p, int n)
{
    int i = blockIdx.x * blockDim.x + threadIdx.x;
    if (i < n) p[i] = 0.f;
}
__global__ void aux_final_kernel(const float* __restrict__ aux, float* __restrict__ outp)
{
    if (threadIdx.x != 0 || blockIdx.x != 0) return;
    float tot = 0.f;
    for (int l = 0; l < L_; ++l) {
        const float* a = aux + l * 2 * E_;
        for (int e = 0; e < E_; ++e) {
            float pm = a[e] / (float)M_;
            float fe = a[E_ + e] / (float)(M_ * TOPK_);
            tot += (float)E_ * fe * pm;
        }
    }
    *outp = tot;
}

// ---------------------------------------------------------------------------
static inline void gemm(hipStream_t st, const float* A, const float* B,
                        const float* bias, const float* scale, int sstride,
                        float* C, int M, int N, int K, int lda, int ldb, int ldc,
                        int transB, int flags,
                        long long sA = 0, long long sB = 0, long long sC = 0,
                        long long sS = 0, int batch = 1)
{
    dim3 grid((N + BN - 1) / BN, (M + BM - 1) / BM, batch);
    gemm_f32_wmma<<<grid, 256, 0, st>>>(A, B, bias, scale, C, M, N, K,
                                        lda, ldb, ldc, sA, sB, sC, sS,
                                        sstride, transB, flags);
}

extern "C" void kernel_launch(void* const* d_in, const int* in_sizes, int n_in,
                              void* d_out, int out_size, void* d_ws, size_t ws_size,
                              hipStream_t stream)
{
    (void)in_sizes; (void)n_in; (void)out_size; (void)ws_size;
    const float* x_in      = (const float*)d_in[0];
    const float* ln1_w     = (const float*)d_in[1];
    const float* ln1_b     = (const float*)d_in[2];
    const float* ln2_w     = (const float*)d_in[3];
    const float* ln2_b     = (const float*)d_in[4];
    const float* qkv_w     = (const float*)d_in[5];
    const float* qkv_b     = (const float*)d_in[6];
    const float* out_w     = (const float*)d_in[7];
    const float* out_b     = (const float*)d_in[8];
    const float* shared_w1 = (const float*)d_in[9];
    const float* shared_b1 = (const float*)d_in[10];
    const float* shared_w2 = (const float*)d_in[11];
    const float* shared_b2 = (const float*)d_in[12];
    const float* gate_w    = (const float*)d_in[13];
    const float* gate_b    = (const float*)d_in[14];
    const float* router_w  = (const float*)d_in[15];
    const float* router_b  = (const float*)d_in[16];
    const float* conv_k    = (const float*)d_in[17];
    const float* conv_w1   = (const float*)d_in[18];
    const float* conv_b1   = (const float*)d_in[19];
    const float* conv_w2   = (const float*)d_in[20];
    const float* conv_b2   = (const float*)d_in[21];
    const float* four_w1   = (const float*)d_in[22];
    const float* four_b1   = (const float*)d_in[23];
    const float* four_w2   = (const float*)d_in[24];
    const float* four_b2   = (const float*)d_in[25];
    const float* mlp_w1    = (const float*)d_in[26];
    const float* mlp_b1    = (const float*)d_in[27];
    const float* mlp_w2    = (const float*)d_in[28];
    const float* mlp_b2    = (const float*)d_in[29];
    const float* log_alpha = (const float*)d_in[30];

    float* xo      = (float*)d_out;                // running x (B,T,D)
    float* aux_out = xo + (size_t)M_ * D_;         // scalar aux

    float* ws     = (float*)d_ws;
    float* normed = ws;  ws += (size_t)M_ * D_;
    float* qkvb   = ws;  ws += (size_t)M_ * 3 * D_;
    float* attno  = ws;  ws += (size_t)M_ * D_;
    float* h1     = ws;  ws += (size_t)M_ * FF_;
    float* outb   = ws;  ws += (size_t)M_ * D_;
    float* cin    = ws;  ws += (size_t)M_ * D_;
    float* gate   = ws;  ws += (size_t)M_ * E_;
    float* sgate  = ws;  ws += (size_t)M_;
    float* freq   = ws;  ws += (size_t)B_ * F_ * 2 * D_;
    float* fo     = ws;  ws += (size_t)B_ * F_ * 2 * D_;
    float* cosF   = ws;  ws += (size_t)F_ * T_;
    float* sinF   = ws;  ws += (size_t)F_ * T_;
    float* icos   = ws;  ws += (size_t)T_ * F_;
    float* isin   = ws;  ws += (size_t)T_ * F_;
    float* auxws  = ws;  ws += (size_t)L_ * 2 * E_;

    const int nMD = M_ * D_;
    copy_kernel<<<(nMD + 255) / 256, 256, 0, stream>>>(x_in, xo, nMD);
    basis_kernel<<<(F_ * T_ + 255) / 256, 256, 0, stream>>>(cosF, sinF, icos, isin);
    zero_kernel<<<1, 64, 0, stream>>>(auxws, L_ * 2 * E_);

    for (int l = 0; l < L_; ++l) {
        // attention block
        ln_kernel<<<M_, 256, 0, stream>>>(xo, ln1_w + l * D_, ln1_b + l * D_, normed);
        gemm(stream, normed, qkv_w + (size_t)l * 3 * D_ * D_, qkv_b + l * 3 * D_,
             nullptr, 0, qkvb, M_, 3 * D_, D_, D_, D_, 3 * D_, 0, 0);
        attn_kernel<<<B_ * H_ * T_ / 8, 256, 0, stream>>>(qkvb, attno, log_alpha,
                                                          (l < 2) ? 1 : 0);
        gemm(stream, attno, out_w + (size_t)l * D_ * D_, out_b + l * D_,
             nullptr, 0, xo, M_, D_, D_, D_, D_, D_, 0, FLAG_ACC);

        // MoE block
        ln_kernel<<<M_, 256, 0, stream>>>(xo, ln2_w + l * D_, ln2_b + l * D_, normed);
        router_kernel<<<(M_ + 255) / 256, 256, 0, stream>>>(normed,
             router_w + (size_t)l * E_ * D_, router_b + l * E_,
             gate_w + (size_t)l * D_, gate_b + l, gate, sgate, auxws + l * 2 * E_);

        // shared expert (sigmoid-gated), overwrites outb
        gemm(stream, normed, shared_w1 + (size_t)l * FF_ * D_, shared_b1 + l * FF_,
             nullptr, 0, h1, M_, FF_, D_, D_, D_, FF_, 0, FLAG_GELU);
        gemm(stream, h1, shared_w2 + (size_t)l * D_ * FF_, shared_b2 + l * D_,
             sgate, 1, outb, M_, D_, FF_, FF_, FF_, D_, 0, FLAG_SCALE);

        for (int s = 0; s < 2; ++s) {
            const size_t ls = (size_t)(l * 2 + s);
            const int ec = s * 3 + 0, ef = s * 3 + 1, em = s * 3 + 2;

            // conv expert
            dwconv_kernel<<<(nMD + 255) / 256, 256, 0, stream>>>(normed,
                conv_k + ls * D_ * 3, cin, nMD);
            gemm(stream, cin, conv_w1 + ls * FF_ * D_, conv_b1 + ls * FF_,
                 nullptr, 0, h1, M_, FF_, D_, D_, D_, FF_, 0, FLAG_GELU);
            gemm(stream, h1, conv_w2 + ls * D_ * FF_, conv_b2 + ls * D_,
                 gate + ec, E_, outb, M_, D_, FF_, FF_, FF_, D_, 0,
                 FLAG_SCALE | FLAG_ACC);

            // fourier expert: rfft as batched DFT GEMMs
            gemm(stream, cosF, normed, nullptr, nullptr, 0, freq,
                 F_, D_, T_, T_, D_, 2 * D_, 1, 0,
                 0, (long long)T_ * D_, (long long)F_ * 2 * D_, 0, B_);
            gemm(stream, sinF, normed, nullptr, nullptr, 0, freq + D_,
                 F_, D_, T_, T_, D_, 2 * D_, 1, 0,
                 0, (long long)T_ * D_, (long long)F_ * 2 * D_, 0, B_);
            gemm(stream, freq, four_w1 + ls * FF_ * 2 * D_, four_b1 + ls * FF_,
                 nullptr, 0, h1, B_ * F_, FF_, 2 * D_, 2 * D_, 2 * D_, FF_, 0,
                 FLAG_GELU);
            gemm(stream, h1, four_w2 + ls * 2 * D_ * FF_, four_b2 + ls * 2 * D_,
                 nullptr, 0, fo, B_ * F_, 2 * D_, FF_, FF_, FF_, 2 * D_, 0, 0);
            // irfft as two batched GEMMs, gated + accumulated into outb
            // (lda=513 is odd -> A staging takes the guarded scalar path)
            gemm(stream, icos, fo, nullptr, gate + ef, E_, outb,
                 T_, D_, F_, F_, 2 * D_, D_, 1, FLAG_SCALE | FLAG_ACC,
                 0, (long long)F_ * 2 * D_, (long long)T_ * D_,
                 (long long)T_ * E_, B_);
            gemm(stream, isin, fo + D_, nullptr, gate + ef, E_, outb,
                 T_, D_, F_, F_, 2 * D_, D_, 1, FLAG_SCALE | FLAG_ACC,
                 0, (long long)F_ * 2 * D_, (long long)T_ * D_,
                 (long long)T_ * E_, B_);

            // mlp expert
            gemm(stream, normed, mlp_w1 + ls * FF_ * D_, mlp_b1 + ls * FF_,
                 nullptr, 0, h1, M_, FF_, D_, D_, D_, FF_, 0, FLAG_GELU);
            gemm(stream, h1, mlp_w2 + ls * D_ * FF_, mlp_b2 + ls * D_,
                 gate + em, E_, outb, M_, D_, FF_, FF_, FF_, D_, 0,
                 FLAG_SCALE | FLAG_ACC);
        }
        add_kernel<<<(nMD + 255) / 256, 256, 0, stream>>>(xo, outb, nMD);
    }
    aux_final_kernel<<<1, 1, 0, stream>>>(auxws, aux_out);
}